// PositionOnlyAttention_87771951661446
// MI455X (gfx1250) — compile-verified
//
#include <hip/hip_runtime.h>
#include <hip/hip_bf16.h>
#include <stdint.h>

#define S 2048
#define TOKEN_BITS 16
#define H 4
#define NPOS 12
#define VBITS 10
#define OBITS 12

#define WAVES_PER_BLOCK 8
#define BLOCK (WAVES_PER_BLOCK * 32)

#define CV_N (H * TOKEN_BITS * VBITS)   // 640 ints
#define CO_N (TOKEN_BITS * OBITS)       // 192 ints

// gfx1250 async global->LDS copy (ASYNCcnt-tracked). VDST = LDS byte offset
// (low 32 bits of a generic shared pointer), VADDR = 64-bit global address.
__device__ __forceinline__ void async_g2l_b32(unsigned lds_off, const void* g) {
    unsigned long long ga = (unsigned long long)(uintptr_t)g;
    asm volatile("global_load_async_to_lds_b32 %0, %1, off"
                 :: "v"(lds_off), "v"(ga) : "memory");
}

__global__ __launch_bounds__(BLOCK) void poa_kernel(
    const int*   __restrict__ tokens,        // (S, 16)        0/1
    const int*   __restrict__ route_table,   // (H, 1<<24)     0/1
    const float* __restrict__ value_table,   // (H, 16, 1024)
    const float* __restrict__ output_table,  // (16, 4096)
    const int*   __restrict__ conns_value,   // (H, 16, 10)    in [0,28)
    const int*   __restrict__ conns_out,     // (16, 12)       in [0,64)
    float*       __restrict__ out)           // (S, 16)
{
    __shared__ int s_cv[CV_N];
    __shared__ int s_co[CO_N];

    // Stage the tiny connection tables into LDS via the gfx1250 async path.
    for (int k = threadIdx.x; k < CV_N; k += BLOCK)
        async_g2l_b32((unsigned)(uintptr_t)&s_cv[k], &conns_value[k]);
    for (int k = threadIdx.x; k < CO_N; k += BLOCK)
        async_g2l_b32((unsigned)(uintptr_t)&s_co[k], &conns_out[k]);
    __builtin_amdgcn_s_wait_asynccnt(0);
    __syncthreads();

    const int lane = threadIdx.x & 31;
    const int wave = threadIdx.x >> 5;
    const int i    = blockIdx.x * WAVES_PER_BLOCK + wave;   // row handled by this wave

    // Issue the first route chunk (j = 0..31) for all 4 heads up front so the
    // four load latencies overlap; with p=0.5 bits, P[chunk empty] = 2^-32.
    int r0[H];
#pragma unroll
    for (int h = 0; h < H; ++h)
        r0[h] = route_table[((size_t)h << (2 * NPOS)) + ((size_t)i << NPOS) + lane];

    unsigned long long combined = 0ull;

#pragma unroll
    for (int h = 0; h < H; ++h) {
        unsigned long long b = __ballot(r0[h] > 0);
        int jfirst;
        if (b) {
            jfirst = (int)__builtin_ctzll(b);
        } else {
            // Rare slow path: continue the scan, 32 j's per iteration.
            jfirst = -1;
            const int* row =
                route_table + (((size_t)h << (2 * NPOS)) + ((size_t)i << NPOS));
            for (int base = 32; base < S; base += 32) {
                int r = row[base + lane];
                b = __ballot(r > 0);
                if (b) { jfirst = base + (int)__builtin_ctzll(b); break; }
            }
        }

        unsigned mask_h = 0u;
        if (jfirst >= 0) {
            // One 64B token row feeds all 10 gathers below.
            __builtin_prefetch(tokens + (size_t)jfirst * TOKEN_BITS, 0, 0);
            bool bit_t = false;
            if (lane < TOKEN_BITS) {
                const int t = lane;
                const int* cv = &s_cv[(h * TOKEN_BITS + t) * VBITS];
                int addr = 0;
#pragma unroll
                for (int m = 0; m < VBITS; ++m) {
                    int c = cv[m];
                    // vin[j][c]: c<16 -> token bit, else MSB-first bit of j.
                    int bit = (c < TOKEN_BITS)
                                  ? (tokens[jfirst * TOKEN_BITS + c] & 1)
                                  : ((jfirst >> (TOKEN_BITS + NPOS - 1 - c)) & 1);
                    addr |= bit << (VBITS - 1 - m);
                }
                float v = value_table[((h * TOKEN_BITS + t) << VBITS) + addr];
                bit_t = (v > 0.5f);
            }
            mask_h = (unsigned)__ballot(bit_t) & 0xFFFFu;
        }
        combined |= (unsigned long long)mask_h << (16 * h);
    }

    // Output stage: lanes 0..15 each produce out[i][t].
    if (lane < TOKEN_BITS) {
        const int t = lane;
        int oaddr = 0;
#pragma unroll
        for (int bpos = 0; bpos < OBITS; ++bpos) {
            int c = s_co[t * OBITS + bpos];
            int bit = (int)((combined >> c) & 1ull);
            oaddr |= bit << (OBITS - 1 - bpos);
        }
        out[i * TOKEN_BITS + t] = output_table[(t << OBITS) + oaddr];
    }
}

extern "C" void kernel_launch(void* const* d_in, const int* in_sizes, int n_in,
                              void* d_out, int out_size, void* d_ws, size_t ws_size,
                              hipStream_t stream) {
    const int*   tokens       = (const int*)d_in[0];
    const int*   route_table  = (const int*)d_in[1];
    const float* value_table  = (const float*)d_in[2];
    const float* output_table = (const float*)d_in[3];
    const int*   conns_value  = (const int*)d_in[4];
    const int*   conns_out    = (const int*)d_in[5];
    float*       out          = (float*)d_out;

    dim3 grid(S / WAVES_PER_BLOCK);   // 256 blocks x 256 threads = 2048 waves
    poa_kernel<<<grid, BLOCK, 0, stream>>>(tokens, route_table, value_table,
                                           output_table, conns_value, conns_out, out);
}